// AffineTransform2D_14602888806510
// MI455X (gfx1250) — compile-verified
//
#include <hip/hip_runtime.h>
#include <math.h>

// MI455X / gfx1250, wave32. One wave per batch; 8 waves (256 threads) per block.
// G = P^T P (5x5 within 16x16) accumulated with V_WMMA_F32_16X16X4_F32 in fp32.

typedef __attribute__((ext_vector_type(2))) float v2f;
typedef __attribute__((ext_vector_type(8))) float v8f;

#define NPTS 512
#define PTS_PER_LANE 16   // 512 points / 32 lanes
#define WAVES_PER_BLOCK 8

__device__ __forceinline__ float wave_sum32(float v) {
#pragma unroll
  for (int m = 16; m >= 1; m >>= 1) v += __shfl_xor(v, m, 32);
  return v;
}

// One Jacobi rotation on symmetric 3x3 S (zeroing S[p][q]); accumulates into Q.
// Called only with literal (p,q,r) so all indexing constant-folds to registers.
__device__ __forceinline__ void jacobi_rot(float S[3][3], float Q[3][3],
                                           int p, int q, int r) {
  float apq = S[p][q];
  float denom = 2.0f * apq;
  float safe = (denom == 0.0f) ? 1e-30f : denom;
  float tau = (S[q][q] - S[p][p]) / safe;
  float t = copysignf(1.0f, tau) / (fabsf(tau) + sqrtf(1.0f + tau * tau));
  t = (apq == 0.0f) ? 0.0f : t;
  float c = 1.0f / sqrtf(1.0f + t * t);
  float s = t * c;
  float Spr = S[p][r], Sqr = S[q][r];
  S[p][r] = S[r][p] = c * Spr - s * Sqr;
  S[q][r] = S[r][q] = s * Spr + c * Sqr;
  S[p][p] -= t * apq;
  S[q][q] += t * apq;
  S[p][q] = S[q][p] = 0.0f;
#pragma unroll
  for (int k = 0; k < 3; ++k) {
    float Qkp = Q[k][p], Qkq = Q[k][q];
    Q[k][p] = c * Qkp - s * Qkq;
    Q[k][q] = s * Qkp + c * Qkq;
  }
}

__global__ __launch_bounds__(256)
void affine2d_wls_kernel(const float* __restrict__ src,
                         const float* __restrict__ dst,
                         const float* __restrict__ wts,
                         float* __restrict__ out, int nbatch) {
  // Per-wave private LDS slice: 32 point-rows of 8 floats (g0..g4, 0,0,0).
  __shared__ float Pbuf[WAVES_PER_BLOCK][32 * 8];

  const int lane = threadIdx.x & 31;
  const int wave = threadIdx.x >> 5;
  const int batch = blockIdx.x * WAVES_PER_BLOCK + wave;
  if (batch >= nbatch) return;  // uniform per wave -> EXEC all-1s at WMMA

  const float2* s2 = (const float2*)src + (size_t)batch * NPTS;
  const float2* d2 = (const float2*)dst + (size_t)batch * NPTS;
  const float*  wp = wts + (size_t)batch * NPTS;

  // ---- Phase 0: one coalesced pass; keep points in registers; moment sums ----
  float sx[PTS_PER_LANE], sy[PTS_PER_LANE], dx[PTS_PER_LANE], dy[PTS_PER_LANE],
      wv[PTS_PER_LANE];
  float ssx = 0.f, ssy = 0.f, ss2 = 0.f, sdx = 0.f, sdy = 0.f, sd2 = 0.f;
#pragma unroll
  for (int j = 0; j < PTS_PER_LANE; ++j) {
    int i = j * 32 + lane;
    float2 s = s2[i];
    float2 d = d2[i];
    sx[j] = s.x; sy[j] = s.y; dx[j] = d.x; dy[j] = d.y;
    wv[j] = wp[i];
    ssx += s.x; ssy += s.y;
    ss2 = fmaf(s.x, s.x, fmaf(s.y, s.y, ss2));
    sdx += d.x; sdy += d.y;
    sd2 = fmaf(d.x, d.x, fmaf(d.y, d.y, sd2));
  }
  ssx = wave_sum32(ssx); ssy = wave_sum32(ssy); ss2 = wave_sum32(ss2);
  sdx = wave_sum32(sdx); sdy = wave_sum32(sdy); sd2 = wave_sum32(sd2);

  const float inv_n = 1.0f / (float)NPTS;
  float csx = ssx * inv_n, csy = ssy * inv_n;
  float cdx = sdx * inv_n, cdy = sdy * inv_n;
  float mss = fmaxf(ss2 * inv_n - (csx * csx + csy * csy), 0.0f);
  float msd = fmaxf(sd2 * inv_n - (cdx * cdx + cdy * cdy), 0.0f);
  float rms_s = sqrtf(mss); rms_s = (rms_s == 0.0f) ? 1.0f : rms_s;
  float rms_d = sqrtf(msd); rms_d = (rms_d == 0.0f) ? 1.0f : rms_d;
  const float SQRT2 = 1.41421356237309515f;
  const float nfs = SQRT2 / rms_s;
  const float nfd = SQRT2 / rms_d;

  // ---- Phase 1: G = P^T P via fp32 WMMA (A and B operands are identical) ----
  float* Pw = &Pbuf[wave][0];
  Pw[lane * 8 + 5] = 0.0f;  // zero pad -> D rows/cols 5..7 stay clean
  Pw[lane * 8 + 6] = 0.0f;
  Pw[lane * 8 + 7] = 0.0f;

  v8f acc = {0.f, 0.f, 0.f, 0.f, 0.f, 0.f, 0.f, 0.f};
  const int half = (lane >> 4) << 1;  // K sub-block: 0 for lanes 0-15, 2 for 16-31
  const int comp = lane & 7;          // row component index (garbage rows >=8 unused)

  for (int c = 0; c < PTS_PER_LANE; ++c) {
    // g = sqrt(w) * [sx', sy', 1, dx', dy'] in normalized coordinates.
    // (w/max(w) scaling dropped: eigenvectors invariant to positive scaling of M.)
    float rw = sqrtf(wv[c]);
    float* row = Pw + lane * 8;
    row[0] = rw * (nfs * (sx[c] - csx));
    row[1] = rw * (nfs * (sy[c] - csy));
    row[2] = rw;
    row[3] = rw * (nfd * (dx[c] - cdx));
    row[4] = rw * (nfd * (dy[c] - cdy));
    // Single wave owns this LDS slice; same-wave DS ops are in-order.
#pragma unroll
    for (int t = 0; t < 8; ++t) {
      int kb = t * 4 + half;
      v2f ab;
      ab.x = Pw[(kb + 0) * 8 + comp];
      ab.y = Pw[(kb + 1) * 8 + comp];
      // D = A*B + C with A = P^T chunk (16x4), B = P chunk (4x16): identical regs.
      acc = __builtin_amdgcn_wmma_f32_16x16x4_f32(false, ab, false, ab,
                                                  (short)0, acc, false, false);
    }
  }

  // ---- Extract 5x5 block of G: lane n (n<16) holds D[m][n] in acc[m] ----
  float G[5][5];
#pragma unroll
  for (int m = 0; m < 5; ++m)
#pragma unroll
    for (int n = 0; n < 5; ++n) G[m][n] = __shfl(acc[m], n, 32);

  // ---- Phase 2 (all lanes redundant, branch-uniform): arrowhead eigen-solve ----
  // M = [[S,0,p],[0,S,q],[p^T,q^T,r]]; diagonalize S = Q L Q^T (3x3 Jacobi).
  float S[3][3], Q[3][3];
#pragma unroll
  for (int i = 0; i < 3; ++i)
#pragma unroll
    for (int j = 0; j < 3; ++j) {
      S[i][j] = G[i][j];
      Q[i][j] = (i == j) ? 1.0f : 0.0f;
    }
#pragma unroll
  for (int sw = 0; sw < 3; ++sw) {
    jacobi_rot(S, Q, 0, 1, 2);
    jacobi_rot(S, Q, 0, 2, 1);
    jacobi_rot(S, Q, 1, 2, 0);
  }
  float lv[3] = {S[0][0], S[1][1], S[2][2]};
  float pv[3] = {G[0][3], G[1][3], G[2][3]};
  float qv[3] = {G[0][4], G[1][4], G[2][4]};
  float rr = G[3][3] + G[4][4];

  float pt[3], qt[3], z2[3];
#pragma unroll
  for (int i = 0; i < 3; ++i) {
    pt[i] = Q[0][i] * pv[0] + Q[1][i] * pv[1] + Q[2][i] * pv[2];
    qt[i] = Q[0][i] * qv[0] + Q[1][i] * qv[1] + Q[2][i] * qv[2];
    z2[i] = pt[i] * pt[i] + qt[i] * qt[i];
  }
  // Secular eq: f(l) = r - l + sum z2_i/(l - L_i); unique root below min(L)
  // is the smallest eigenvalue of M (f strictly decreasing there).
  float lmin = fminf(lv[0], fminf(lv[1], lv[2]));
  float znorm = sqrtf(z2[0] + z2[1] + z2[2]);
  float lo = fminf(lmin, rr) - znorm - 1.0f;  // f(lo) > 0 (Gershgorin-type bound)
  float hi = lmin;                            // f(hi^-) -> -inf
  for (int it = 0; it < 48; ++it) {
    float mid = 0.5f * (lo + hi);
    float f = rr - mid;
#pragma unroll
    for (int i = 0; i < 3; ++i) f += z2[i] / (mid - lv[i]);
    bool pos = f > 0.0f;
    lo = pos ? mid : lo;
    hi = pos ? hi : mid;
  }
  float lam = 0.5f * (lo + hi);

  // Eigenvector (c = v[6] = 1): a_i~ = p_i~/(lam-L_i); back-rotate by Q.
  float at[3], bt[3];
#pragma unroll
  for (int i = 0; i < 3; ++i) {
    float inv = 1.0f / (lam - lv[i]);
    at[i] = pt[i] * inv;
    bt[i] = qt[i] * inv;
  }
  float a[3], b[3];
#pragma unroll
  for (int k = 0; k < 3; ++k) {
    a[k] = Q[k][0] * at[0] + Q[k][1] * at[1] + Q[k][2] * at[2];
    b[k] = Q[k][0] * bt[0] + Q[k][1] * bt[1] + Q[k][2] * bt[2];
  }

  // ---- Denormalize: H = Td^-1 * Hn * Ts (bottom rows are [0,0,1]) ----
  float h00 = -a[0], h01 = -a[1], h02 = -a[2];
  float h10 = -b[0], h11 = -b[1], h12 = -b[2];
  float X00 = h00 * nfs, X01 = h01 * nfs;
  float X02 = -nfs * (h00 * csx + h01 * csy) + h02;
  float X10 = h10 * nfs, X11 = h11 * nfs;
  float X12 = -nfs * (h10 * csx + h11 * csy) + h12;
  float invfd = 1.0f / nfd;
  float H00 = X00 * invfd, H01 = X01 * invfd, H02 = X02 * invfd + cdx;
  float H10 = X10 * invfd, H11 = X11 * invfd, H12 = X12 * invfd + cdy;
  // H[2][2] == 1 exactly by construction, so the final normalization is identity.

  if (lane == 0) {
    float* o = out + (size_t)batch * 9;
    o[0] = H00; o[1] = H01; o[2] = H02;
    o[3] = H10; o[4] = H11; o[5] = H12;
    o[6] = 0.0f; o[7] = 0.0f; o[8] = 1.0f;
  }
}

extern "C" void kernel_launch(void* const* d_in, const int* in_sizes, int n_in,
                              void* d_out, int out_size, void* d_ws, size_t ws_size,
                              hipStream_t stream) {
  const float* src = (const float*)d_in[0];
  const float* dst = (const float*)d_in[1];
  const float* wts = (const float*)d_in[2];
  float* out = (float*)d_out;
  int nbatch = in_sizes[2] / NPTS;  // weights: (B, N)
  dim3 block(32 * WAVES_PER_BLOCK);
  dim3 grid((nbatch + WAVES_PER_BLOCK - 1) / WAVES_PER_BLOCK);
  affine2d_wls_kernel<<<grid, block, 0, stream>>>(src, dst, wts, out, nbatch);
}